// DynamicTopologyCoupler_28355374088772
// MI455X (gfx1250) — compile-verified
//
#include <hip/hip_runtime.h>
#include <hip/hip_bf16.h>

typedef __attribute__((ext_vector_type(2))) float v2f;
typedef __attribute__((ext_vector_type(8))) float v8f;

#define DEV __device__ __forceinline__

DEV float gelu_f(float x) { return 0.5f * x * (1.0f + erff(x * 0.70710678118654752f)); }
DEV float sigm_f(float x) { return 1.0f / (1.0f + expf(-x)); }

// ---------------------------------------------------------------------------
// Generic fp32 WMMA GEMM:  C[M,N] = act(rowmap(A)[M,K] @ B[K,N] + bias) * post
//   - rowmap(m) = (m / seg_rows) * seg_stride_rows + (m % seg_rows)
//   - BTRANS (compile-time): B element [k][n] read as W[n*ldw + k] (nf @ nf^T)
//   - ACT (compile-time): 0 = none, 1 = gelu
//   - post_sig: if non-null, multiply result by sigmoid(*post_sig)
//   - batched over blockIdx.z with element strides sA/sW/sC
// Block = (32,4): 4 waves, each owning one 16x16 tile; grid=(M/16, N/64, batch)
// Inner loop is branch-free pointer-bump: 2 loads + v_wmma per K-step.
// ---------------------------------------------------------------------------
template <int BTRANS, int ACT>
__global__ void wmma_gemm_f32(const float* __restrict__ A, const float* __restrict__ W,
                              const float* __restrict__ bias, float* __restrict__ C,
                              int M, int N, int K, int lda, int ldw, int ldc, int ccol,
                              int seg_rows, int seg_stride_rows,
                              const float* __restrict__ post_sig,
                              long sA, long sW, long sC) {
    A += (long)blockIdx.z * sA;
    W += (long)blockIdx.z * sW;
    C += (long)blockIdx.z * sC;
    const int mt = blockIdx.x * 16;
    const int nt = blockIdx.y * 64 + threadIdx.y * 16;
    if (mt >= M || nt >= N) return;            // uniform per wave -> EXEC all 1s
    const int lane = threadIdx.x;              // 0..31 (wave32)
    const int lo = lane & 15;
    const int hi = lane >> 4;
    const int hi2 = hi << 1;                   // 0 or 2  (A/B K-offset per ISA layout)
    const int m = mt + lo;
    const int arow = (m / seg_rows) * seg_stride_rows + (m % seg_rows);
    const int n = nt + lo;

    const float* ap = A + (long)arow * lda + hi2;
    const float* bp = BTRANS ? (W + (long)n * ldw + hi2)
                             : (W + (long)hi2 * ldw + n);
    const long bstep = BTRANS ? 4 : 4 * (long)ldw;

    v8f acc = {};
#pragma unroll 4
    for (int k0 = 0; k0 < K; k0 += 4) {
        v2f a;
        a.x = ap[0];
        a.y = ap[1];
        ap += 4;
        v2f b;
        if (BTRANS) {
            b.x = bp[0];
            b.y = bp[1];
        } else {
            b.x = bp[0];
            b.y = bp[ldw];
        }
        bp += bstep;
        acc = __builtin_amdgcn_wmma_f32_16x16x4_f32(false, a, false, b, (short)0, acc,
                                                    false, false);
    }

    const float bv = bias ? bias[n] : 0.0f;
    const float ps = post_sig ? sigm_f(*post_sig) : 1.0f;
    const int rbase = hi * 8;                  // C/D layout: hi lanes hold rows +8
    float* crow = C + (long)(mt + rbase) * ldc + ccol + n;
#pragma unroll
    for (int r = 0; r < 8; ++r) {
        float v = acc[r] + bv;
        if (ACT == 1) v = gelu_f(v);
        v *= ps;
        crow[(long)r * ldc] = v;
    }
}

// ---------------------------------------------------------------------------
// Row LayerNorm (+ optional gelu), one block (256 thr) per row; out can carry
// a column offset / ldc so results land directly inside concat buffers.
// ---------------------------------------------------------------------------
__global__ void ln_act_kernel(const float* __restrict__ X, const float* __restrict__ g,
                              const float* __restrict__ b, float* __restrict__ Y,
                              int N, int ldx, int ldy, int ycol, int act) {
    const long row = blockIdx.x;
    const float* x = X + row * ldx;
    __shared__ float red[256];
    const int tid = threadIdx.x;
    float s = 0.0f, s2 = 0.0f;
    for (int c = tid; c < N; c += 256) { float v = x[c]; s += v; s2 += v * v; }
    red[tid] = s; __syncthreads();
    for (int o = 128; o > 0; o >>= 1) { if (tid < o) red[tid] += red[tid + o]; __syncthreads(); }
    const float mean = red[0] / (float)N;
    __syncthreads();
    red[tid] = s2; __syncthreads();
    for (int o = 128; o > 0; o >>= 1) { if (tid < o) red[tid] += red[tid + o]; __syncthreads(); }
    const float var = red[0] / (float)N - mean * mean;
    const float inv = rsqrtf(var + 1e-5f);
    for (int c = tid; c < N; c += 256) {
        float v = (x[c] - mean) * inv * g[c] + b[c];
        if (act == 1) v = gelu_f(v);
        Y[row * ldy + ycol + c] = v;
    }
}

// ---------------------------------------------------------------------------
// RBF-kernel entropy -> coupling strength cs.  One block per (b,i) row.
// sq_j comes from the Gram diagonal; two-pass (sum, then entropy) via LDS.
// ---------------------------------------------------------------------------
__global__ void entropy_cs_kernel(const float* __restrict__ G, const float* __restrict__ temp,
                                  const float* __restrict__ tgt, float* __restrict__ cs,
                                  int Nseq) {
    const int row = blockIdx.x;                // 0..B*Nseq-1
    const int bb = row / Nseq, i = row % Nseq;
    const float* gbat = G + (long)bb * Nseq * Nseq;
    const float* grow = gbat + (long)i * Nseq;
    __shared__ float kv[512];
    __shared__ float red[256];
    const int tid = threadIdx.x;
    const float t = *temp;
    const float inv2t2 = 1.0f / (2.0f * t * t);
    const float sqi = gbat[(long)i * Nseq + i];
    float s = 0.0f;
    for (int j = tid; j < Nseq; j += 256) {
        const float sqj = gbat[(long)j * Nseq + j];
        const float d2 = fmaxf(sqi + sqj - 2.0f * grow[j], 0.0f);
        const float k = __expf(-d2 * inv2t2);
        kv[j] = k; s += k;
    }
    red[tid] = s; __syncthreads();
    for (int o = 128; o > 0; o >>= 1) { if (tid < o) red[tid] += red[tid + o]; __syncthreads(); }
    const float invS = 1.0f / red[0];
    __syncthreads();
    float e = 0.0f;
    for (int j = tid; j < Nseq; j += 256) {
        const float p = kv[j] * invS;
        e -= p * __logf(p + 1e-6f);
    }
    red[tid] = e; __syncthreads();
    for (int o = 128; o > 0; o >>= 1) { if (tid < o) red[tid] += red[tid + o]; __syncthreads(); }
    if (tid == 0) cs[row] = sigm_f(-(red[0] - *tgt) / t);
}

// ---------------------------------------------------------------------------
// Fused per-(b,h) attention, 16 query rows per block (1 wave).
// S=QK^T via WMMA into LDS, wave32 shuffle softmax, O=PV via WMMA.
// Dims baked in: Nseq=512, D=256, H=8, hd=32 -> fully unrolled QK^T K-loop.
// ---------------------------------------------------------------------------
__global__ void attention_kernel(const float* __restrict__ Q, const float* __restrict__ K,
                                 const float* __restrict__ V, float* __restrict__ O) {
    constexpr int Nseq = 512, D = 256, hd = 32;
    const int mt = blockIdx.x * 16;
    const int h = blockIdx.y;
    const int bb = blockIdx.z;
    const int lane = threadIdx.x;
    const int lo = lane & 15;
    const int hi = lane >> 4;
    const int hi2 = hi << 1;
    __shared__ float S[16][Nseq];
    const float scale = rsqrtf((float)hd);
    const long base = (long)bb * Nseq * D + (long)h * hd;

    // S tile (16 x Nseq)
    const float* qrow = Q + base + (long)(mt + lo) * D + hi2;
    for (int ntile = 0; ntile < Nseq; ntile += 16) {
        v8f acc = {};
        const float* krow = K + base + (long)(ntile + lo) * D + hi2;
#pragma unroll
        for (int k0 = 0; k0 < hd; k0 += 4) {
            v2f a; a.x = qrow[k0]; a.y = qrow[k0 + 1];
            v2f b; b.x = krow[k0]; b.y = krow[k0 + 1];
            acc = __builtin_amdgcn_wmma_f32_16x16x4_f32(false, a, false, b, (short)0, acc,
                                                        false, false);
        }
#pragma unroll
        for (int r = 0; r < 8; ++r) S[r + hi * 8][ntile + lo] = acc[r] * scale;
    }
    __syncthreads();

    // softmax over each of the 16 rows (wave32 shuffle reductions)
    for (int r = 0; r < 16; ++r) {
        float mx = -1e30f;
        for (int c = lane; c < Nseq; c += 32) mx = fmaxf(mx, S[r][c]);
        for (int o = 16; o > 0; o >>= 1) mx = fmaxf(mx, __shfl_xor(mx, o, 32));
        float sum = 0.0f;
        for (int c = lane; c < Nseq; c += 32) { float e = __expf(S[r][c] - mx); S[r][c] = e; sum += e; }
        for (int o = 16; o > 0; o >>= 1) sum += __shfl_xor(sum, o, 32);
        const float inv = 1.0f / sum;
        for (int c = lane; c < Nseq; c += 32) S[r][c] *= inv;
    }
    __syncthreads();

    // O = P @ V  (16 x hd)
#pragma unroll
    for (int nt2 = 0; nt2 < hd; nt2 += 16) {
        v8f acc = {};
        const float* vp = V + base + (long)hi2 * D + nt2 + lo;
#pragma unroll 4
        for (int k0 = 0; k0 < Nseq; k0 += 4) {
            v2f a; a.x = S[lo][k0 + hi2]; a.y = S[lo][k0 + hi2 + 1];
            v2f b; b.x = vp[0]; b.y = vp[D];
            vp += 4 * D;
            acc = __builtin_amdgcn_wmma_f32_16x16x4_f32(false, a, false, b, (short)0, acc,
                                                        false, false);
        }
#pragma unroll
        for (int r = 0; r < 8; ++r)
            O[base + (long)(mt + r + hi * 8) * D + nt2 + lo] = acc[r];
    }
}

// ---------------------------------------------------------------------------
// Tiny-N GEMM heads (N in {1,2}) with wave reduction + sigmoid epilogue.
// ---------------------------------------------------------------------------
__global__ void small_gemm(const float* __restrict__ A, const float* __restrict__ W,
                           const float* __restrict__ bias, float* __restrict__ C,
                           int N, int K, int lda, int act) {
    const int row = blockIdx.x;
    const int lane = threadIdx.x;
    const float* a = A + (long)row * lda;
    for (int j = 0; j < N; ++j) {
        float s = 0.0f;
        for (int k = lane; k < K; k += 32) s += a[k] * W[(long)k * N + j];
        for (int o = 16; o > 0; o >>= 1) s += __shfl_xor(s, o, 32);
        if (lane == 0) {
            float v = s + bias[j];
            if (act == 2) v = sigm_f(v);
            else if (act == 1) v = gelu_f(v);
            C[(long)row * N + j] = v;
        }
    }
}

// ---------------------------------------------------------------------------
// Elementwise glue
// ---------------------------------------------------------------------------
__global__ void scale_cs_kernel(const float* __restrict__ nf, const float* __restrict__ cs,
                                float* __restrict__ cf) {
    const int idx = blockIdx.x * 256 + threadIdx.x;     // 1024*256
    cf[idx] = nf[idx] * cs[idx >> 8];
}

__global__ void build_lc_kernel(const float* __restrict__ ttit, const float* __restrict__ flow,
                                const float* __restrict__ cs, float* __restrict__ lc) {
    const int idx = blockIdx.x * 256 + threadIdx.x;     // 512*256
    const int r = idx >> 8, c = idx & 255;
    const int bb = r >> 8, i = r & 255;
    const float tt = ttit[r * 512 + c], it = ttit[r * 512 + 256 + c];
    const float tf = tt * flow[r * 2], imf = it * flow[r * 2 + 1];
    const float e = cs[bb * 512 + i];
    lc[r * 768 + c] = tf;
    lc[r * 768 + 256 + c] = imf;
    lc[r * 768 + 512 + c] = e * (tf + imf);
}

__global__ void build_x2_kernel(const float* __restrict__ ttit, const float* __restrict__ flow,
                                const float* __restrict__ li, float* __restrict__ x2) {
    const int idx = blockIdx.x * 256 + threadIdx.x;     // 512*256
    const int r = idx >> 8, c = idx & 255;
    const int bb = r >> 8, i = r & 255;
    const float tt = ttit[r * 512 + c], it = ttit[r * 512 + 256 + c];
    const float tf = tt * flow[r * 2], imf = it * flow[r * 2 + 1];
    const float l = li[r];
    x2[(long)(bb * 512 + i) * 256 + c] = tt + l * imf;
    x2[(long)(bb * 512 + 256 + i) * 256 + c] = it + l * tf;
}

__global__ void build_x2cat_kernel(const float* __restrict__ x2, const float* __restrict__ cs,
                                   float* __restrict__ xc) {
    const int idx = blockIdx.x * 256 + threadIdx.x;     // 1024*256
    const int q = idx >> 8, c = idx & 255;
    const float v = x2[idx];
    xc[q * 512 + c] = v;
    xc[q * 512 + 256 + c] = cs[q] * v;
}

__global__ void fuse_kernel(const float* __restrict__ topoenh, const float* __restrict__ gate,
                            float* __restrict__ fused) {
    const int idx = blockIdx.x * 256 + threadIdx.x;     // 1024*256
    const int q = idx >> 8, c = idx & 255;
    const float g = gate[q];
    fused[idx] = g * topoenh[q * 512 + c] + (1.0f - g) * topoenh[q * 512 + 256 + c];
}

// cadj[b,i,j,0] = (1.0 > threshold) * cs[b,i]  (edge_probs == softmax over size-1
// axis == 1 exactly); also emits the cs output.
__global__ void cadj_cs_kernel(const float* __restrict__ cs, const float* __restrict__ thr,
                               float* __restrict__ out) {
    const long idx = (long)blockIdx.x * 256 + threadIdx.x;   // 2*512*512
    const int bb = (int)(idx / (512 * 512));
    const int i = (int)((idx / 512) % 512);
    const float adj = (1.0f > *thr) ? 1.0f : 0.0f;
    out[262144 + idx] = adj * cs[bb * 512 + i];
    if (idx < 1024) out[262144 + 524288 + idx] = cs[idx];
}

// ---------------------------------------------------------------------------
extern "C" void kernel_launch(void* const* d_in, const int* in_sizes, int n_in,
                              void* d_out, int out_size, void* d_ws, size_t ws_size,
                              hipStream_t stream) {
    (void)in_sizes; (void)n_in; (void)out_size; (void)ws_size;
    const float* x      = (const float*)d_in[0];
    const float* pm1w   = (const float*)d_in[1];  const float* pm1b = (const float*)d_in[2];
    const float* pmlng  = (const float*)d_in[3];  const float* pmlnb = (const float*)d_in[4];
    const float* pm2w   = (const float*)d_in[5];  const float* pm2b = (const float*)d_in[6];
    const float* ne1w   = (const float*)d_in[7];  const float* ne1b = (const float*)d_in[8];
    const float* nelng  = (const float*)d_in[9];  const float* nelnb = (const float*)d_in[10];
    const float* ne2w   = (const float*)d_in[11]; const float* ne2b = (const float*)d_in[12];
    const float* thrp   = (const float*)d_in[17];
    const float* tgtp   = (const float*)d_in[24];
    const float* tempp  = (const float*)d_in[25];
    const float* wqw    = (const float*)d_in[26]; const float* wqb = (const float*)d_in[27];
    const float* wkw    = (const float*)d_in[28]; const float* wkb = (const float*)d_in[29];
    const float* wvw    = (const float*)d_in[30]; const float* wvb = (const float*)d_in[31];
    const float* wow    = (const float*)d_in[32]; const float* wob = (const float*)d_in[33];
    const float* at1w   = (const float*)d_in[34]; const float* at1b = (const float*)d_in[35];
    const float* atlng  = (const float*)d_in[36]; const float* atlnb = (const float*)d_in[37];
    const float* ai1w   = (const float*)d_in[38]; const float* ai1b = (const float*)d_in[39];
    const float* ailng  = (const float*)d_in[40]; const float* ailnb = (const float*)d_in[41];
    const float* af1w   = (const float*)d_in[42]; const float* af1b = (const float*)d_in[43];
    const float* aflng  = (const float*)d_in[44]; const float* aflnb = (const float*)d_in[45];
    const float* af2w   = (const float*)d_in[46]; const float* af2b = (const float*)d_in[47];
    const float* al1w   = (const float*)d_in[48]; const float* al1b = (const float*)d_in[49];
    const float* allng  = (const float*)d_in[50]; const float* allnb = (const float*)d_in[51];
    const float* al2w   = (const float*)d_in[52]; const float* al2b = (const float*)d_in[53];
    const float* tt1w   = (const float*)d_in[54]; const float* tt1b = (const float*)d_in[55];
    const float* ttlng  = (const float*)d_in[56]; const float* ttlnb = (const float*)d_in[57];
    const float* ea1w   = (const float*)d_in[58]; const float* ea1b = (const float*)d_in[59];
    const float* ealng  = (const float*)d_in[60]; const float* ealnb = (const float*)d_in[61];
    const float* fg1w   = (const float*)d_in[62]; const float* fg1b = (const float*)d_in[63];
    const float* cc1w   = (const float*)d_in[64]; const float* cc1b = (const float*)d_in[65];
    const float* orderp = (const float*)d_in[66];

    float* outF = (float*)d_out;
    float* ws = (float*)d_ws;

    // workspace bump allocation (elements)
    float* t0    = ws;                  // 1024x512, reused: pm1out -> ne1out -> x2cat -> topo|enh
    float* phase = ws + 524288;         // 1024x256
    float* nf    = ws + 786432;         // 1024x256
    float* G     = ws + 1048576;        // 2x512x512
    float* cs    = ws + 1572864;        // 1024
    float* cf    = ws + 1573888;        // 1024x256
    float* qb    = ws + 1836032;        // 1024x256
    float* kb    = ws + 2098176;        // 1024x256
    float* vb    = ws + 2360320;        // 1024x256
    float* obuf  = ws + 2622464;        // 1024x256
    float* mo    = ws + 2884608;        // 1024x256
    float* tmpA  = ws + 3146752;        // up to 1024x256
    float* tmpB  = ws + 3408896;        // up to 1024x256
    float* ttit  = ws + 3670784;        // 512x512 (tt | it)
    float* flow  = ws + 3932928;        // 512x2
    float* lc    = ws + 3933952;        // 512x768
    float* altmp = ws + 4327168;        // 512x256
    float* li    = ws + 4458240;        // 512
    float* aftmp = ws + 4458752;        // 512x256
    float* x2    = ws + 4589824;        // 1024x256
    float* gate  = ws + 4851968;        // 1024
    float* fused = ws + 4852992;        // 1024x256

    const dim3 blk(32, 4);

    // phase = lin(gelu(ln(lin(x, pm1), pm_ln)), pm2)
    wmma_gemm_f32<0, 0><<<dim3(64, 8, 1), blk, 0, stream>>>(x, pm1w, pm1b, t0, 1024, 512, 256,
        256, 512, 512, 0, 1024, 0, nullptr, 0, 0, 0);
    ln_act_kernel<<<1024, 256, 0, stream>>>(t0, pmlng, pmlnb, t0, 512, 512, 512, 0, 1);
    wmma_gemm_f32<0, 0><<<dim3(64, 4, 1), blk, 0, stream>>>(t0, pm2w, pm2b, phase, 1024, 256, 512,
        512, 256, 256, 0, 1024, 0, nullptr, 0, 0, 0);

    // nf = lin(gelu(ln(lin(phase, ne1), ne_ln)), ne2)
    wmma_gemm_f32<0, 0><<<dim3(64, 8, 1), blk, 0, stream>>>(phase, ne1w, ne1b, t0, 1024, 512, 256,
        256, 512, 512, 0, 1024, 0, nullptr, 0, 0, 0);
    ln_act_kernel<<<1024, 256, 0, stream>>>(t0, nelng, nelnb, t0, 512, 512, 512, 0, 1);
    wmma_gemm_f32<0, 0><<<dim3(64, 4, 1), blk, 0, stream>>>(t0, ne2w, ne2b, nf, 1024, 256, 512,
        512, 256, 256, 0, 1024, 0, nullptr, 0, 0, 0);

    // Gram matrix G[b] = nf[b] @ nf[b]^T (batched, B-transposed)
    wmma_gemm_f32<1, 0><<<dim3(32, 8, 2), blk, 0, stream>>>(nf, nf, nullptr, G, 512, 512, 256,
        256, 256, 512, 0, 512, 0, nullptr, 131072, 131072, 262144);

    // entropy -> cs; cf = nf * cs
    entropy_cs_kernel<<<1024, 256, 0, stream>>>(G, tempp, tgtp, cs, 512);
    scale_cs_kernel<<<1024, 256, 0, stream>>>(nf, cs, cf);

    // attention
    wmma_gemm_f32<0, 0><<<dim3(64, 4, 1), blk, 0, stream>>>(cf, wqw, wqb, qb, 1024, 256, 256,
        256, 256, 256, 0, 1024, 0, nullptr, 0, 0, 0);
    wmma_gemm_f32<0, 0><<<dim3(64, 4, 1), blk, 0, stream>>>(cf, wkw, wkb, kb, 1024, 256, 256,
        256, 256, 256, 0, 1024, 0, nullptr, 0, 0, 0);
    wmma_gemm_f32<0, 0><<<dim3(64, 4, 1), blk, 0, stream>>>(cf, wvw, wvb, vb, 1024, 256, 256,
        256, 256, 256, 0, 1024, 0, nullptr, 0, 0, 0);
    attention_kernel<<<dim3(32, 8, 2), 32, 0, stream>>>(qb, kb, vb, obuf);
    wmma_gemm_f32<0, 0><<<dim3(64, 4, 1), blk, 0, stream>>>(obuf, wow, wob, mo, 1024, 256, 256,
        256, 256, 256, 0, 1024, 0, nullptr, 0, 0, 0);

    // tt / it branches on sequence halves (row remapping over mo)
    wmma_gemm_f32<0, 0><<<dim3(32, 4, 1), blk, 0, stream>>>(mo, at1w, at1b, tmpA, 512, 256, 256,
        256, 256, 256, 0, 256, 512, nullptr, 0, 0, 0);
    wmma_gemm_f32<0, 0><<<dim3(32, 4, 1), blk, 0, stream>>>(mo + 256 * 256, ai1w, ai1b, tmpB,
        512, 256, 256, 256, 256, 256, 0, 256, 512, nullptr, 0, 0, 0);
    ln_act_kernel<<<512, 256, 0, stream>>>(tmpA, atlng, atlnb, ttit, 256, 256, 512, 0, 1);
    ln_act_kernel<<<512, 256, 0, stream>>>(tmpB, ailng, ailnb, ttit, 256, 256, 512, 256, 1);

    // flow
    wmma_gemm_f32<0, 0><<<dim3(32, 4, 1), blk, 0, stream>>>(ttit, af1w, af1b, aftmp, 512, 256, 512,
        512, 256, 256, 0, 512, 0, nullptr, 0, 0, 0);
    ln_act_kernel<<<512, 256, 0, stream>>>(aftmp, aflng, aflnb, aftmp, 256, 256, 256, 0, 1);
    small_gemm<<<512, 32, 0, stream>>>(aftmp, af2w, af2b, flow, 2, 256, 256, 2);

    // li
    build_lc_kernel<<<512, 256, 0, stream>>>(ttit, flow, cs, lc);
    wmma_gemm_f32<0, 0><<<dim3(32, 4, 1), blk, 0, stream>>>(lc, al1w, al1b, altmp, 512, 256, 768,
        768, 256, 256, 0, 512, 0, nullptr, 0, 0, 0);
    ln_act_kernel<<<512, 256, 0, stream>>>(altmp, allng, allnb, altmp, 256, 256, 256, 0, 1);
    small_gemm<<<512, 32, 0, stream>>>(altmp, al2w, al2b, li, 1, 256, 256, 2);

    // x2, topo, enh
    build_x2_kernel<<<512, 256, 0, stream>>>(ttit, flow, li, x2);
    wmma_gemm_f32<0, 0><<<dim3(64, 4, 1), blk, 0, stream>>>(x2, tt1w, tt1b, tmpA, 1024, 256, 256,
        256, 256, 256, 0, 1024, 0, nullptr, 0, 0, 0);
    build_x2cat_kernel<<<1024, 256, 0, stream>>>(x2, cs, t0);
    wmma_gemm_f32<0, 0><<<dim3(64, 4, 1), blk, 0, stream>>>(t0, ea1w, ea1b, tmpB, 1024, 256, 512,
        512, 256, 256, 0, 1024, 0, nullptr, 0, 0, 0);
    ln_act_kernel<<<1024, 256, 0, stream>>>(tmpA, ttlng, ttlnb, t0, 256, 256, 512, 0, 1);
    ln_act_kernel<<<1024, 256, 0, stream>>>(tmpB, ealng, ealnb, t0, 256, 256, 512, 256, 1);

    // gate, fused, crit
    small_gemm<<<1024, 32, 0, stream>>>(t0, fg1w, fg1b, gate, 1, 512, 512, 2);
    fuse_kernel<<<1024, 256, 0, stream>>>(t0, gate, fused);
    wmma_gemm_f32<0, 1><<<dim3(64, 4, 1), blk, 0, stream>>>(fused, cc1w, cc1b, outF, 1024, 256, 256,
        256, 256, 256, 0, 1024, 0, orderp, 0, 0, 0);

    // cadj + cs outputs
    cadj_cs_kernel<<<2048, 256, 0, stream>>>(cs, thrp, outF);
}